// SlotFillingCoachModel_19275813224817
// MI455X (gfx1250) — compile-verified
//
#include <hip/hip_runtime.h>
#include <hip/hip_bf16.h>

#define BS     64
#define SEQ    1024
#define DDIM   800
#define NSLOT  64
#define NLAB   72
#define SLOTS_PER_BLK 16
#define LDS_STRIDE 804   // pad 800 -> 804 so 16 A-rows hit 16 distinct LDS banks
#define NTILE  5         // ceil(72/16) N-tiles
#define KCHUNK (DDIM / 4) // 200 WMMA k-steps

typedef float v2f __attribute__((ext_vector_type(2)));
typedef float v8f __attribute__((ext_vector_type(8)));

// ---------------------------------------------------------------------------
// Prep: build B fragments in the exact per-lane V_WMMA_F32_16X16X4_F32 layout,
// zero-padded to 80 columns. frag[((t*200 + c)*32 + lane)] = {B[kk][n], B[kk+1][n]}
// with kk = 4c + (lane>>4)*2, n = 16t + (lane&15). 256 KB total, L2-resident.
// ---------------------------------------------------------------------------
__global__ __launch_bounds__(256)
void build_wfrag_kernel(const float* __restrict__ W, float2* __restrict__ frag)
{
    const int idx = blockIdx.x * blockDim.x + threadIdx.x;
    if (idx >= NTILE * KCHUNK * 32) return;
    const int lane = idx & 31;
    const int c    = (idx >> 5) % KCHUNK;
    const int t    = idx / (KCHUNK * 32);
    const int kk   = 4 * c + (lane >> 4) * 2;
    const int n    = 16 * t + (lane & 15);
    float2 v = make_float2(0.f, 0.f);
    if (n < NLAB) {
        v.x = W[(size_t)kk * NLAB + n];
        v.y = W[(size_t)(kk + 1) * NLAB + n];
    }
    frag[idx] = v;
}

// ---------------------------------------------------------------------------
// Main fused kernel: ragged mean-pool 16 slots into LDS, then
// (16 x 800) @ (800 x 72) via v_wmma_f32_16x16x4_f32.
// ---------------------------------------------------------------------------
template <bool USE_FRAG>
__global__ __launch_bounds__(256)
void slot_pool_clf_kernel(const float* __restrict__ hs,
                          const int*   __restrict__ begins,
                          const int*   __restrict__ lens,
                          const float* __restrict__ W,
                          const float* __restrict__ bias,
                          const float2* __restrict__ wfrag,
                          float*       __restrict__ out)
{
    __shared__ float pool[SLOTS_PER_BLK * LDS_STRIDE];  // ~51.5 KB

    const int b     = blockIdx.x >> 2;                  // batch
    const int slot0 = (blockIdx.x & 3) * SLOTS_PER_BLK; // first slot of this tile
    const int tid   = threadIdx.x;

    // ---- Phase 1: ragged mean-pool, coalesced global_load_b128 per span row ----
    for (int s = 0; s < SLOTS_PER_BLK; ++s) {
        const int gi    = b * NSLOT + slot0 + s;        // uniform -> scalar loads
        const int begin = begins[gi];
        const int len   = lens[gi];                     // span = len+1 rows (<= 8)
        const float inv = 1.0f / (float)(len + 1);
        const int c4 = tid;
        if (c4 < DDIM / 4) {
            const float4* row = (const float4*)hs
                              + (size_t)(b * SEQ + begin) * (DDIM / 4) + c4;
            float4 acc = make_float4(0.f, 0.f, 0.f, 0.f);
            for (int r = 0; r <= len; ++r) {
                float4 v = row[(size_t)r * (DDIM / 4)];
                acc.x += v.x; acc.y += v.y; acc.z += v.z; acc.w += v.w;
            }
            float* dst = &pool[s * LDS_STRIDE + c4 * 4];
            dst[0] = acc.x * inv; dst[1] = acc.y * inv;
            dst[2] = acc.z * inv; dst[3] = acc.w * inv;
        }
    }
    __syncthreads();

    // ---- Phase 2: WMMA GEMM, wave w (0..4) owns one 16-wide N-tile ----
    const int w = tid >> 5;           // wave id (wave32)
    if (w < NTILE) {
        const int lane = tid & 31;
        const int m    = lane & 15;   // A row (slot) / output label col within tile
        const int grp  = lane >> 4;   // 0: K={0,1}, 1: K={2,3} per fp32 A layout
        const int n    = w * 16 + m;  // output label column
        const bool valid = (n < NLAB);

        const float*  arow = &pool[m * LDS_STRIDE + grp * 2];
        const float2* brow = USE_FRAG
                           ? (wfrag + (size_t)w * KCHUNK * 32 + lane)
                           : nullptr;
        v8f acc = {};
        #pragma unroll 4
        for (int c = 0; c < KCHUNK; ++c) {
            const int k = 4 * c;
            v2f a, bb;
            a.x = arow[k];            // ds_load_b64: A[m][k + grp*2 .. +1]
            a.y = arow[k + 1];
            if (USE_FRAG) {
                float2 bv = brow[(size_t)c * 32];  // unconditional global_load_b64
                bb.x = bv.x;
                bb.y = bv.y;
            } else {
                const int kk = k + grp * 2;
                bb.x = valid ? W[(size_t)kk * NLAB + n] : 0.0f;
                bb.y = valid ? W[(size_t)(kk + 1) * NLAB + n] : 0.0f;
            }
            acc = __builtin_amdgcn_wmma_f32_16x16x4_f32(
                      /*neg_a=*/false, a, /*neg_b=*/false, bb,
                      /*c_mod=*/(short)0, acc,
                      /*reuse_a=*/false, /*reuse_b=*/false);
        }

        if (valid) {
            const float bn = bias[n];
            const size_t obase = (size_t)(b * NSLOT + slot0) * NLAB;
            #pragma unroll
            for (int r = 0; r < 8; ++r) {
                const int mrow = r + grp * 8;   // C/D layout: VGPR r -> M=r (+8 hi half)
                out[obase + (size_t)mrow * NLAB + n] = acc[r] + bn;
            }
        }
    }
}

extern "C" void kernel_launch(void* const* d_in, const int* in_sizes, int n_in,
                              void* d_out, int out_size, void* d_ws, size_t ws_size,
                              hipStream_t stream) {
    const float* hs     = (const float*)d_in[0];  // (64,1024,800) f32
    const int*   begins = (const int*)  d_in[1];  // (64,64) i32
    const int*   lens   = (const int*)  d_in[2];  // (64,64) i32
    const float* W      = (const float*)d_in[3];  // (800,72) f32
    const float* bias   = (const float*)d_in[4];  // (72,) f32
    float*       out    = (float*)d_out;          // (64,64,72) f32

    constexpr size_t FRAG_ELEMS = (size_t)NTILE * KCHUNK * 32;      // float2 count
    constexpr size_t FRAG_BYTES = FRAG_ELEMS * sizeof(float2);      // 256 000 B

    dim3 grid(BS * (NSLOT / SLOTS_PER_BLK));      // 256 blocks
    dim3 block(256);                              // 8 wave32 waves

    if (d_ws != nullptr && ws_size >= FRAG_BYTES) {
        float2* frag = (float2*)d_ws;
        const int prep_threads = (int)FRAG_ELEMS; // 32 000
        build_wfrag_kernel<<<(prep_threads + 255) / 256, 256, 0, stream>>>(W, frag);
        slot_pool_clf_kernel<true><<<grid, block, 0, stream>>>(
            hs, begins, lens, W, bias, frag, out);
    } else {
        slot_pool_clf_kernel<false><<<grid, block, 0, stream>>>(
            hs, begins, lens, W, bias, nullptr, out);
    }
}